// ConnectomeDecoder_53575422050534
// MI455X (gfx1250) — compile-verified
//
#include <hip/hip_runtime.h>

typedef __attribute__((ext_vector_type(2))) float v2f;
typedef __attribute__((ext_vector_type(8))) float v8f;

#define Cn   64
#define Tn   256
#define Bn   8
#define H1n  32
#define H2n  64
#define CHn  4096            // H2n * Cn
#define RDIM 12288           // CHn * 3 (ic-major, tap fastest: matches conv_w inner layout)
#define KSPLIT 4
#define KCHUNK 3072          // RDIM / KSPLIT

// workspace layout in floats
#define WS_A    0                        // 64*64
#define WS_HB   4096                     // HbigT: 16 * 12288 (rows 8..15 stay zero)
#define WS_PART (4096 + 16*RDIM)         // partial: KSPLIT * 8 * 4096

__device__ __forceinline__ float relu_(float v) { return fmaxf(v, 0.0f); }

// ---------------- Kernel 1: normalized adjacency ----------------
__global__ __launch_bounds__(64) void prep_adj_kernel(const float* __restrict__ adj,
                                                      float* __restrict__ Aout) {
  __shared__ float dinv[Cn];
  int i = threadIdx.x;
  float deg = 0.0f;
  for (int j = 0; j < Cn; ++j) deg += adj[i * Cn + j];
  dinv[i] = rsqrtf(fmaxf(deg, 1.0f));
  __syncthreads();
  float di = dinv[i];
  for (int j = 0; j < Cn; ++j)
    Aout[i * Cn + j] = di * adj[i * Cn + j] * dinv[j];
}

// ---------------- Kernel 2: zero the activation matrix ----------------
__global__ __launch_bounds__(256) void zero_kernel(float* __restrict__ p, int n) {
  int idx = blockIdx.x * 256 + threadIdx.x;
  if (idx < n) p[idx] = 0.0f;
}

// ---------------- Kernel 3: GCN at the 3 needed time steps ----------------
// blockIdx.x = b*3 + ktap ; t = 247 + 4*ktap. 64 threads, thread i = node index.
__global__ __launch_bounds__(64) void gcn_kernel(const float* __restrict__ x,
                                                 const float* __restrict__ A,
                                                 const float* __restrict__ W1,
                                                 const float* __restrict__ b1,
                                                 const float* __restrict__ W2,
                                                 const float* __restrict__ b2,
                                                 float* __restrict__ HbigT) {
  __shared__ float As[Cn * Cn];      // 16 KB
  __shared__ float W2s[H2n * H1n];   // 8 KB
  __shared__ float h1s[Cn * H1n];    // 8 KB
  __shared__ float xv[Cn], W1s[H1n], b1s[H1n], b2s[H2n];

  const int b = blockIdx.x / 3;
  const int ktap = blockIdx.x % 3;
  const int t = 247 + 4 * ktap;
  const int tid = threadIdx.x;

  for (int idx = tid; idx < Cn * Cn; idx += 64) As[idx] = A[idx];
  for (int idx = tid; idx < H2n * H1n; idx += 64) W2s[idx] = W2[idx];
  if (tid < H1n) { W1s[tid] = W1[tid]; b1s[tid] = b1[tid]; }
  b2s[tid] = b2[tid];
  xv[tid] = x[(size_t)(b * Cn + tid) * Tn + t];
  __syncthreads();

  const int i = tid;
  // layer 1: s1 = (A @ x), h1 = relu(s1 * W1^T + b1)
  float s1 = 0.0f;
  for (int j = 0; j < Cn; ++j) s1 += As[i * Cn + j] * xv[j];
  for (int f = 0; f < H1n; ++f) h1s[i * H1n + f] = relu_(s1 * W1s[f] + b1s[f]);
  __syncthreads();

  // layer 2: s2 = A @ h1 ; h2 = relu(s2 @ W2^T + b2)
  float s2[H1n];
  for (int f = 0; f < H1n; ++f) s2[f] = 0.0f;
  for (int j = 0; j < Cn; ++j) {
    float a = As[i * Cn + j];
    const float* h1r = &h1s[j * H1n];
    for (int f = 0; f < H1n; ++f) s2[f] += a * h1r[f];
  }
  float* dst = HbigT + (size_t)b * RDIM + (size_t)i * H2n * 3 + ktap;
  for (int j2 = 0; j2 < H2n; ++j2) {
    const float* w2r = &W2s[j2 * H1n];
    float v = b2s[j2];
    for (int f = 0; f < H1n; ++f) v += s2[f] * w2r[f];
    dst[j2 * 3] = relu_(v);  // r = (i*64 + j2)*3 + ktap, column b
  }
}

// ---------------- Kernel 4: main GEMM via V_WMMA_F32_16X16X4_F32 ----------------
// grid = (256, KSPLIT), block = 32 (one wave). M tile = 16 channels, N = 16 (8 real batches).
__global__ __launch_bounds__(32) void conv_wmma_kernel(const float* __restrict__ conv_w,
                                                       const float* __restrict__ HbigT,
                                                       float* __restrict__ partial) {
  const int ch0  = blockIdx.x * 16;
  const int p    = blockIdx.y;
  const int lane = threadIdx.x;        // 0..31
  const int half = lane >> 4;          // lanes 16..31 hold K+2,K+3
  const int lmod = lane & 15;

  const int r0 = p * KCHUNK;
  // A fragment source: row (ch0+lmod) of conv_w, K offset r0 + 2*half
  const float* aptr = conv_w + (size_t)(ch0 + lmod) * RDIM + r0 + 2 * half;
  // B fragment source: column n=lmod of H (stored transposed [n][r]); rows 8..15 are zero
  const float* bptr = HbigT + (size_t)lmod * RDIM + r0 + 2 * half;

  v8f acc = {0.f, 0.f, 0.f, 0.f, 0.f, 0.f, 0.f, 0.f};
  for (int k = 0; k < KCHUNK; k += 4) {
    v2f af = *(const v2f*)(aptr + k);
    v2f bf = *(const v2f*)(bptr + k);
    acc = __builtin_amdgcn_wmma_f32_16x16x4_f32(
        /*neg_a=*/false, af, /*neg_b=*/false, bf,
        /*c_mod=*/(short)0, acc, /*reuse_a=*/false, /*reuse_b=*/false);
  }

  // D layout: lane L -> n = L&15 (batch), VGPR v -> m = 8*half + v (channel)
  const int n = lmod;
  if (n < Bn) {
    float* dst = partial + ((size_t)(p * Bn + n) * CHn) + ch0 + 8 * half;
    for (int v = 0; v < 8; ++v) dst[v] = acc[v];
  }
}

// ---------------- Kernel 5: reduce partials, bias+relu, head GEMV ----------------
__global__ __launch_bounds__(128) void head_kernel(const float* __restrict__ partial,
                                                   const float* __restrict__ conv_b,
                                                   const float* __restrict__ head_w,
                                                   const float* __restrict__ head_b,
                                                   float* __restrict__ out) {
  const int tid = threadIdx.x;   // 0..127
  const int b = tid >> 4;
  const int l = tid & 15;
  float acc = head_b[l];
  for (int ch = 0; ch < CHn; ++ch) {
    float y = conv_b[ch];
    for (int p = 0; p < KSPLIT; ++p)
      y += partial[((size_t)(p * Bn + b) * CHn) + ch];
    y = relu_(y);
    acc += head_w[l * CHn + ch] * y;
  }
  out[b * 16 + l] = acc;
}

extern "C" void kernel_launch(void* const* d_in, const int* in_sizes, int n_in,
                              void* d_out, int out_size, void* d_ws, size_t ws_size,
                              hipStream_t stream) {
  const float* x      = (const float*)d_in[0];
  const float* adj    = (const float*)d_in[1];
  const float* W1     = (const float*)d_in[2];
  const float* b1     = (const float*)d_in[3];
  const float* W2     = (const float*)d_in[4];
  const float* b2     = (const float*)d_in[5];
  const float* conv_w = (const float*)d_in[6];
  const float* conv_b = (const float*)d_in[7];
  const float* head_w = (const float*)d_in[8];
  const float* head_b = (const float*)d_in[9];
  float* out = (float*)d_out;
  float* ws  = (float*)d_ws;

  float* A      = ws + WS_A;
  float* HbigT  = ws + WS_HB;
  float* part   = ws + WS_PART;

  prep_adj_kernel<<<1, 64, 0, stream>>>(adj, A);

  const int hbN = 16 * RDIM;
  zero_kernel<<<(hbN + 255) / 256, 256, 0, stream>>>(HbigT, hbN);

  gcn_kernel<<<Bn * 3, 64, 0, stream>>>(x, A, W1, b1, W2, b2, HbigT);

  conv_wmma_kernel<<<dim3(CHn / 16, KSPLIT), 32, 0, stream>>>(conv_w, HbigT, part);

  head_kernel<<<1, 128, 0, stream>>>(part, conv_b, head_w, head_b, out);
}